// PMFM_55542517072500
// MI455X (gfx1250) — compile-verified
//
#include <hip/hip_runtime.h>

typedef float v2f __attribute__((ext_vector_type(2)));
typedef float v4f __attribute__((ext_vector_type(4)));
typedef float v8f __attribute__((ext_vector_type(8)));

#define THETA   0.5f
#define EPS     1e-5f
#define LOG2E   1.4426950408889634f

// ---------------------------------------------------------------------------
// K1: per 64x64 patch, compute sum(x) and sum(x^2) using the CDNA5 fp32 WMMA
// path: D = ones(16x4) * B(4x16) + C accumulates column sums of data tiles.
// Grid: one 256-thread WG (8 wave32) per patch; 8192 WGs total.
// ---------------------------------------------------------------------------
__global__ __launch_bounds__(256) void pmfm_patch_reduce(
    const float* __restrict__ x, float* __restrict__ sums, float* __restrict__ sqs)
{
    const int gid   = blockIdx.x;          // plane*16 + patch
    const int plane = gid >> 4;
    const int patch = gid & 15;
    const int py = patch >> 2, px = patch & 3;
    const float* base = x + (size_t)plane * 65536u + (size_t)(py * 64) * 256u + (size_t)(px * 64);

    const int tid  = threadIdx.x;
    const int wave = tid >> 5;             // 0..7 -> rows [8w, 8w+8)
    const int lane = tid & 31;
    const int half = (lane >> 4) & 1;      // B layout: lanes 16..31 hold K=2,3
    const int n    = lane & 15;            // column within 16-wide group

    v8f csum = {0.f,0.f,0.f,0.f,0.f,0.f,0.f,0.f};
    v8f csq  = {0.f,0.f,0.f,0.f,0.f,0.f,0.f,0.f};
    const v2f ones = {1.0f, 1.0f};         // A = 16x4 all-ones (2 VGPRs)

    const float* wbase = base + wave * 8 * 256;
#pragma unroll
    for (int rg = 0; rg < 2; ++rg) {       // row groups of 4 (K dimension)
#pragma unroll
        for (int cg = 0; cg < 4; ++cg) {   // column groups of 16 (N dimension)
            // B tile 4x16: v0 = rows {rg*4, rg*4+2}, v1 = rows {rg*4+1, rg*4+3}
            const float* t = wbase + (rg * 4 + 2 * half) * 256 + cg * 16 + n;
            v2f b;
            b[0] = t[0];
            b[1] = t[256];
            v2f b2 = b * b;
            csum = __builtin_amdgcn_wmma_f32_16x16x4_f32(
                false, ones, false, b,  (short)0, csum, false, false);
            csq  = __builtin_amdgcn_wmma_f32_16x16x4_f32(
                false, ones, false, b2, (short)0, csq,  false, false);
        }
    }

    // Every C row holds identical column sums -> sum all 256 slots, divide by 16.
    float s = csum[0]+csum[1]+csum[2]+csum[3]+csum[4]+csum[5]+csum[6]+csum[7];
    float q = csq[0] +csq[1] +csq[2] +csq[3] +csq[4] +csq[5] +csq[6] +csq[7];
#pragma unroll
    for (int m = 16; m >= 1; m >>= 1) {
        s += __shfl_xor(s, m, 32);
        q += __shfl_xor(q, m, 32);
    }
    s *= 0.0625f;
    q *= 0.0625f;

    __shared__ float ls[8], lq[8];
    if (lane == 0) { ls[wave] = s; lq[wave] = q; }
    __syncthreads();
    if (tid == 0) {
        float S = 0.f, Q = 0.f;
#pragma unroll
        for (int w = 0; w < 8; ++w) { S += ls[w]; Q += lq[w]; }
        sums[gid] = S;
        sqs[gid]  = Q;
    }
}

// ---------------------------------------------------------------------------
// K2: per patch, fold patch stats into the 3 (mu, inv_std) pairs relevant to
// this patch (p=2: itself, p=1: its quadrant, p=0: whole plane), then apply
//   out = x * ( theta*sum(lw) + (1-theta) * sum_i lw_i * sigmoid((x-mu_i)*inv_i) )
// sigmoid via single-instruction v_exp_f32 / v_rcp_f32:
//   sigmoid((x-mu)*inv) = rcp(1 + exp2(b - g*x)),  g = inv*log2e, b = mu*g
// x reads hit L2 (resident from K1); output uses non-temporal b128 stores.
// ---------------------------------------------------------------------------
__global__ __launch_bounds__(256) void pmfm_apply(
    const float* __restrict__ x, const float* __restrict__ lvlw,
    const float* __restrict__ sums, const float* __restrict__ sqs,
    float* __restrict__ out)
{
    const int gid   = blockIdx.x;
    const int plane = gid >> 4;
    const int patch = gid & 15;
    const int py = patch >> 2, px = patch & 3;

    __shared__ float sh[10];
    if (threadIdx.x == 0) {
        const float* S = sums + plane * 16;
        const float* Q = sqs  + plane * 16;
        // level p=2: this 64x64 patch
        float s2 = S[patch], q2 = Q[patch];
        // level p=1: 128x128 quadrant containing this patch
        const int qy = py & ~1, qx = px & ~1;
        float s1 = 0.f, q1 = 0.f;
#pragma unroll
        for (int dy = 0; dy < 2; ++dy)
#pragma unroll
            for (int dx = 0; dx < 2; ++dx) {
                int pp = (qy + dy) * 4 + (qx + dx);
                s1 += S[pp]; q1 += Q[pp];
            }
        // level p=0: whole plane
        float s0 = 0.f, q0 = 0.f;
#pragma unroll
        for (int pp = 0; pp < 16; ++pp) { s0 += S[pp]; q0 += Q[pp]; }

        float mu2 = s2 * (1.f / 4096.f);
        float mu1 = s1 * (1.f / 16384.f);
        float mu0 = s0 * (1.f / 65536.f);
        float i2 = rsqrtf(q2 * (1.f / 4096.f)  - mu2 * mu2 + EPS);
        float i1 = rsqrtf(q1 * (1.f / 16384.f) - mu1 * mu1 + EPS);
        float i0 = rsqrtf(q0 * (1.f / 65536.f) - mu0 * mu0 + EPS);

        // softmax over the 3 level weights; feats order is p=2, p=1, p=0
        float w0 = lvlw[0], w1 = lvlw[1], w2 = lvlw[2];
        float mx = fmaxf(w0, fmaxf(w1, w2));
        float e0 = __expf(w0 - mx), e1 = __expf(w1 - mx), e2 = __expf(w2 - mx);
        float r  = 1.f / (e0 + e1 + e2);
        float lwA = e0 * r, lwB = e1 * r, lwC = e2 * r;   // p=2, p=1, p=0

        // precompute exp2-domain sigmoid coefficients
        float g2 = i2 * LOG2E, g1 = i1 * LOG2E, g0 = i0 * LOG2E;
        sh[0] = g2; sh[1] = mu2 * g2;
        sh[2] = g1; sh[3] = mu1 * g1;
        sh[4] = g0; sh[5] = mu0 * g0;
        sh[6] = THETA * (lwA + lwB + lwC);
        sh[7] = (1.f - THETA) * lwA;
        sh[8] = (1.f - THETA) * lwB;
        sh[9] = (1.f - THETA) * lwC;
    }
    __syncthreads();

    const float g2 = sh[0], b2 = sh[1];
    const float g1 = sh[2], b1 = sh[3];
    const float g0 = sh[4], b0 = sh[5];
    const float a  = sh[6], c2 = sh[7], c1 = sh[8], c0 = sh[9];

    const size_t off = (size_t)plane * 65536u + (size_t)(py * 64) * 256u + (size_t)(px * 64);
    const float* xb = x + off;
    float*       ob = out + off;
    const int tid = threadIdx.x;

#pragma unroll
    for (int i = 0; i < 4; ++i) {
        int idx = i * 256 + tid;          // 1024 float4s per patch
        int r = idx >> 4;
        int c = (idx & 15) << 2;
        const v4f v = *(const v4f*)(xb + r * 256 + c);
        v4f o;
#pragma unroll
        for (int k = 0; k < 4; ++k) {
            float xv = v[k];
            // sigmoid((x-mu)*inv) = 1 / (1 + 2^(b - g*x))
            float sA = __builtin_amdgcn_rcpf(1.f + __builtin_amdgcn_exp2f(__builtin_fmaf(-g2, xv, b2)));
            float sB = __builtin_amdgcn_rcpf(1.f + __builtin_amdgcn_exp2f(__builtin_fmaf(-g1, xv, b1)));
            float sC = __builtin_amdgcn_rcpf(1.f + __builtin_amdgcn_exp2f(__builtin_fmaf(-g0, xv, b0)));
            o[k] = xv * (a + c2 * sA + c1 * sB + c0 * sC);
        }
        __builtin_nontemporal_store(o, (v4f*)(ob + r * 256 + c));
    }
}

extern "C" void kernel_launch(void* const* d_in, const int* in_sizes, int n_in,
                              void* d_out, int out_size, void* d_ws, size_t ws_size,
                              hipStream_t stream) {
    const float* x   = (const float*)d_in[0];
    const float* lw  = (const float*)d_in[1];
    float*       out = (float*)d_out;

    const int planes = in_sizes[0] / 65536;   // B*C = 512 for the reference shape
    const int npatch = planes * 16;           // 8192 64x64 patches

    float* sums = (float*)d_ws;               // npatch floats
    float* sqs  = sums + npatch;              // npatch floats (total 64 KB)

    pmfm_patch_reduce<<<npatch, 256, 0, stream>>>(x, sums, sqs);
    pmfm_apply<<<npatch, 256, 0, stream>>>(x, lw, sums, sqs, out);
}